// MultiLevelRoIGenerator_32719060861127
// MI455X (gfx1250) — compile-verified
//
#include <hip/hip_runtime.h>
#include <stdint.h>

#define BATCH    8
#define NLVL     5
#define CAND     2048      // candidate buffer per (batch,level)
#define NCAT     4768      // 1000*4 + 768 concatenated proposals
#define FINAL_K  1000
#define BBOX_CLIP 4.135166556742356f   // log(1000/16)
#define NMS_THR  0.7f

#define AS1 __attribute__((address_space(1)))
#define AS3 __attribute__((address_space(3)))

typedef int v4i __attribute__((ext_vector_type(4)));

#if defined(__has_builtin)
#  if __has_builtin(__builtin_amdgcn_global_load_async_to_lds_b128)
#    define HAVE_ASYNC_LDS 1
#  endif
#endif

// ---------- helpers ----------

// monotonic float->uint key (handles negatives); top-k by logit == top-k by sigmoid
__device__ __forceinline__ unsigned fkey(float x) {
  unsigned u = __float_as_uint(x);
  return (u >> 31) ? ~u : (u | 0x80000000u);
}

// strict total order: key descending, index ascending on ties (matches lax.top_k)
__device__ __forceinline__ bool pair_gt(unsigned ka, int ia, unsigned kb, int ib) {
  return (ka > kb) || (ka == kb && ia < ib);
}

// in-LDS bitonic sort, descending by (key, idx-asc). n power of two.
__device__ inline void bitonic_desc(unsigned* keys, int* vals, int n, int tid, int bd) {
  for (int k2 = 2; k2 <= n; k2 <<= 1) {
    for (int j = k2 >> 1; j > 0; j >>= 1) {
      __syncthreads();
      for (int i = tid; i < n; i += bd) {
        int l = i ^ j;
        if (l > i) {
          unsigned ka = keys[i], kb = keys[l];
          int ia = vals[i], ib = vals[l];
          bool descBlock = ((i & k2) == 0);
          bool sw = descBlock ? pair_gt(kb, ib, ka, ia) : pair_gt(ka, ia, kb, ib);
          if (sw) { keys[i] = kb; vals[i] = ib; keys[l] = ka; vals[l] = ia; }
        }
      }
    }
  }
  __syncthreads();
}

// ---------- top-k body: radix threshold select + bitonic sort ----------
// writes kk sorted-descending indices (into the nb-length array s) to selOut.
__device__ void topk_body(const float* __restrict__ s, int nb, int kk,
                          int* __restrict__ selOut,
                          unsigned* hist, unsigned* keys, int* idxs,
                          unsigned* sh_prefix, int* sh_kth, unsigned* sh_cnt,
                          int tid, int bd) {
  unsigned T = 0;
  if (nb > CAND) {
    if (tid == 0) { *sh_prefix = 0u; *sh_kth = kk; }
    __syncthreads();
    for (int p = 0; p < 4; ++p) {
      for (int i = tid; i < 256; i += bd) hist[i] = 0u;
      __syncthreads();
      const int shift = 24 - 8 * p;
      const unsigned pref = *sh_prefix;
      for (int i = tid; i < nb; i += bd) {
        unsigned key = fkey(s[i]);
        if (p == 0 || (key >> (shift + 8)) == pref)
          atomicAdd(&hist[(key >> shift) & 255u], 1u);
      }
      __syncthreads();
      if (tid == 0) {
        int kth = *sh_kth; unsigned cum = 0; int bsel = 0;
        for (int bin = 255; bin >= 0; --bin) {
          unsigned h = hist[bin];
          if ((unsigned)kth <= cum + h) { bsel = bin; *sh_kth = kth - (int)cum; break; }
          cum += h;
        }
        *sh_prefix = (pref << 8) | (unsigned)bsel;
      }
      __syncthreads();
    }
    T = *sh_prefix;   // exact key of the kk-th largest element
  }

  // deterministic-set compaction: all strictly-greater first, then equals fill up
  if (tid == 0) *sh_cnt = 0u;
  __syncthreads();
  for (int i = tid; i < nb; i += bd) {
    unsigned key = fkey(s[i]);
    if (key > T) {
      unsigned p = atomicAdd(sh_cnt, 1u);
      if (p < CAND) { keys[p] = key; idxs[p] = i; }
    }
  }
  __syncthreads();
  for (int i = tid; i < nb; i += bd) {
    unsigned key = fkey(s[i]);
    if (key == T) {
      unsigned p = atomicAdd(sh_cnt, 1u);
      if (p < CAND) { keys[p] = key; idxs[p] = i; }
    }
  }
  __syncthreads();
  unsigned total = *sh_cnt; if (total > CAND) total = CAND;
  for (int i = tid; i < CAND; i += bd)
    if ((unsigned)i >= total) { keys[i] = 0u; idxs[i] = 0x7FFFFFFF; }
  __syncthreads();

  bitonic_desc(keys, idxs, CAND, tid, bd);

  for (int r = tid; r < kk; r += bd)
    selOut[r] = idxs[r];
}

// ---------- fused per-level parameter blocks (kernel args by value) ----------
struct SelParams {
  const float* scores[NLVL];
  int*         selIdx[NLVL];
  int          nb[NLVL];
  int          kk[NLVL];
};
struct DecParams {
  const float* enc[NLVL];
  const float* logits[NLVL];
  const float* anchors[NLVL];
  const int*   selIdx[NLVL];
  float*       selB[NLVL];
  float*       selS[NLVL];
  const float* img;
  int          nb[NLVL];
  int          kk[NLVL];
  int          tstart[NLVL];
  int          ttotal;
};
struct NmsParams {
  const float* selB[NLVL];
  const float* selS[NLVL];
  float*       allB;
  float*       allS;
  int          kk[NLVL];
  int          post[NLVL];
  int          off[NLVL];
};

// ---------- kernel 1: selection for all (level,batch) pairs in one launch ----------
__global__ __launch_bounds__(1024) void select_topk_fused(SelParams P) {
  __shared__ unsigned hist[256];
  __shared__ unsigned keys[CAND];
  __shared__ int      idxs[CAND];
  __shared__ unsigned sh_prefix;
  __shared__ int      sh_kth;
  __shared__ unsigned sh_cnt;
  const int lvl = blockIdx.x >> 3;
  const int bb  = blockIdx.x & 7;
  const int nb = P.nb[lvl], kk = P.kk[lvl];
  topk_body(P.scores[lvl] + (size_t)bb * nb, nb, kk,
            P.selIdx[lvl] + (size_t)bb * CAND,
            hist, keys, idxs, &sh_prefix, &sh_kth, &sh_cnt,
            threadIdx.x, blockDim.x);
}

// same body, single-array version for the final cross-level top-1000
__global__ __launch_bounds__(1024) void select_topk_single(
    const float* __restrict__ scores, int nb, int kk, int* __restrict__ selIdx) {
  __shared__ unsigned hist[256];
  __shared__ unsigned keys[CAND];
  __shared__ int      idxs[CAND];
  __shared__ unsigned sh_prefix;
  __shared__ int      sh_kth;
  __shared__ unsigned sh_cnt;
  const int bb = blockIdx.x;
  topk_body(scores + (size_t)bb * nb, nb, kk, selIdx + (size_t)bb * CAND,
            hist, keys, idxs, &sh_prefix, &sh_kth, &sh_cnt,
            threadIdx.x, blockDim.x);
}

// ---------- kernel 2: gather + box decode + clip + sigmoid (all levels) ----------
__global__ void decode_fused(DecParams P) {
  int t = blockIdx.x * blockDim.x + threadIdx.x;
  if (t >= P.ttotal) return;
  int lvl = NLVL - 1;
  for (int l = 0; l < NLVL - 1; ++l)
    if (t < P.tstart[l + 1]) { lvl = l; break; }
  const int nb = P.nb[lvl], kk = P.kk[lvl];
  int local = t - P.tstart[lvl];
  int bb = local / kk, r = local - bb * kk;

  int a = P.selIdx[lvl][(size_t)bb * CAND + r];
  size_t base = ((size_t)bb * nb + a) * 4;
  const float* enc = P.enc[lvl];
  const float* anc = P.anchors[lvl];

  float dy = enc[base + 0], dx = enc[base + 1];
  float dh = fminf(enc[base + 2], BBOX_CLIP);
  float dw = fminf(enc[base + 3], BBOX_CLIP);
  float ay0 = anc[base + 0], ax0 = anc[base + 1];
  float ay1 = anc[base + 2], ax1 = anc[base + 3];
  float ah = ay1 - ay0, aw = ax1 - ax0;
  float ayc = ay0 + 0.5f * ah, axc = ax0 + 0.5f * aw;
  float yc = dy * ah + ayc, xc = dx * aw + axc;
  float h = expf(dh) * ah, w = expf(dw) * aw;
  float y0 = yc - 0.5f * h, x0 = xc - 0.5f * w;
  float y1 = yc + 0.5f * h, x1 = xc + 0.5f * w;
  float H = P.img[bb * 2 + 0], W = P.img[bb * 2 + 1];
  y0 = fminf(fmaxf(y0, 0.f), H); x0 = fminf(fmaxf(x0, 0.f), W);
  y1 = fminf(fmaxf(y1, 0.f), H); x1 = fminf(fmaxf(x1, 0.f), W);

  size_t o = ((size_t)bb * CAND + r) * 4;
  float* outB = P.selB[lvl];
  outB[o + 0] = y0; outB[o + 1] = x0; outB[o + 2] = y1; outB[o + 3] = x1;
  float lg = P.logits[lvl][(size_t)bb * nb + a];
  P.selS[lvl][(size_t)bb * CAND + r] = 1.f / (1.f + expf(-lg));
}

// ---------- kernel 3: greedy sorted NMS in LDS, all (level,batch) in one launch ----
// boxes staged via CDNA5 async global->LDS (b128, ASYNCcnt)
__global__ __launch_bounds__(512) void nms_fused(NmsParams P) {
  const int lvl = blockIdx.x >> 3;
  const int bb  = blockIdx.x & 7;
  const int tid = threadIdx.x, bd = blockDim.x;
  const int kk = P.kk[lvl], post_k = P.post[lvl], lvlOff = P.off[lvl];

  __shared__ float4 box[2000];
  __shared__ float  area[2000];
  __shared__ int    kept[2000];
  __shared__ int    pos[2000];
  __shared__ int    sh_total;

  const float4* gsrc = (const float4*)(P.selB[lvl] + (size_t)bb * CAND * 4);
  const float*  selS = P.selS[lvl];

#if defined(HAVE_ASYNC_LDS)
  // gfx1250 async path: per-lane 16B global->LDS DMA, tracked by ASYNCcnt
  for (int e = tid; e < kk; e += bd) {
    __builtin_amdgcn_global_load_async_to_lds_b128(
        (AS1 v4i*)(gsrc + e), (AS3 v4i*)(&box[e]), 0, 0);
  }
#  if __has_builtin(__builtin_amdgcn_s_wait_asynccnt)
  __builtin_amdgcn_s_wait_asynccnt(0);
#  else
  asm volatile("s_wait_asynccnt 0" ::: "memory");
#  endif
#else
  for (int e = tid; e < kk; e += bd) box[e] = gsrc[e];
#endif
  for (int e = tid; e < kk; e += bd) kept[e] = 1;
  __syncthreads();
  for (int e = tid; e < kk; e += bd) {
    float4 v = box[e];
    area[e] = (v.z - v.x) * (v.w - v.y);
  }
  __syncthreads();

  // greedy suppression over sorted-descending candidates
  for (int i = 0; i < kk; ++i) {
    if (kept[i]) {
      float4 bi = box[i];
      float  ai = area[i];
      for (int j = i + 1 + tid; j < kk; j += bd) {
        if (kept[j]) {
          float4 bj = box[j];
          float iy = fmaxf(0.f, fminf(bi.z, bj.z) - fmaxf(bi.x, bj.x));
          float ix = fmaxf(0.f, fminf(bi.w, bj.w) - fmaxf(bi.y, bj.y));
          float inter = iy * ix;
          float iou = inter / (ai + area[j] - inter + 1e-8f);
          if (iou > NMS_THR) kept[j] = 0;
        }
      }
    }
    __syncthreads();
  }

  // ordered compaction of survivors; zero-pad to post_k (matches ref top_k of zeroed arrays)
  if (tid == 0) {
    int c = 0;
    for (int j = 0; j < kk; ++j) if (kept[j]) pos[j] = c++;
    sh_total = c;
  }
  __syncthreads();
  const int total = sh_total;
  float* aS = P.allS + (size_t)bb * NCAT + lvlOff;
  float* aB = P.allB + ((size_t)bb * NCAT + lvlOff) * 4;
  for (int j = tid; j < kk; j += bd) {
    if (kept[j] && pos[j] < post_k) {
      int p = pos[j];
      aS[p] = selS[(size_t)bb * CAND + j];
      float4 v = box[j];
      aB[p * 4 + 0] = v.x; aB[p * 4 + 1] = v.y;
      aB[p * 4 + 2] = v.z; aB[p * 4 + 3] = v.w;
    }
  }
  for (int p = total + tid; p < post_k; p += bd) {
    aS[p] = 0.f;
    aB[p * 4 + 0] = 0.f; aB[p * 4 + 1] = 0.f;
    aB[p * 4 + 2] = 0.f; aB[p * 4 + 3] = 0.f;
  }
}

// ---------- kernel 4: final gather into d_out ----------
__global__ void final_gather_kernel(
    const float* __restrict__ allB, const float* __restrict__ allS,
    const int* __restrict__ selIdx, float* __restrict__ outRois,
    float* __restrict__ outScores) {
  int t = blockIdx.x * blockDim.x + threadIdx.x;
  if (t >= BATCH * FINAL_K) return;
  int bb = t / FINAL_K, r = t - bb * FINAL_K;
  int idx = selIdx[(size_t)bb * CAND + r];
  const float* src = allB + ((size_t)bb * NCAT + idx) * 4;
  float* o = outRois + (size_t)t * 4;
  o[0] = src[0]; o[1] = src[1]; o[2] = src[2]; o[3] = src[3];
  outScores[t] = allS[(size_t)bb * NCAT + idx];
}

// ---------- host ----------
extern "C" void kernel_launch(void* const* d_in, const int* in_sizes, int n_in,
                              void* d_out, int out_size, void* d_ws, size_t ws_size,
                              hipStream_t stream) {
  (void)in_sizes; (void)n_in; (void)out_size; (void)ws_size;

  static const int NB[NLVL]  = {196608, 49152, 12288, 3072, 768};
  static const int KK[NLVL]  = {2000, 2000, 2000, 2000, 768};
  static const int PK[NLVL]  = {1000, 1000, 1000, 1000, 768};
  static const int OFF[NLVL] = {0, 1000, 2000, 3000, 4000};

  char* ws = (char*)d_ws;
  size_t off = 0;
  auto alloc = [&](size_t bytes) -> void* {
    void* p = ws + off;
    off = (off + bytes + 255) & ~(size_t)255;
    return p;
  };

  SelParams sp; DecParams dp; NmsParams np;
  int tcur = 0;
  for (int l = 0; l < NLVL; ++l) {
    dp.enc[l]     = (const float*)d_in[3 * l + 0];
    sp.scores[l]  = (const float*)d_in[3 * l + 1];
    dp.logits[l]  = sp.scores[l];
    dp.anchors[l] = (const float*)d_in[3 * l + 2];
    sp.nb[l] = dp.nb[l] = NB[l];
    sp.kk[l] = dp.kk[l] = np.kk[l] = KK[l];
    np.post[l] = PK[l];
    np.off[l]  = OFF[l];
    dp.tstart[l] = tcur;
    tcur += BATCH * KK[l];
  }
  dp.ttotal = tcur;
  dp.img = (const float*)d_in[15];

  for (int l = 0; l < NLVL; ++l) {
    int* si = (int*)alloc((size_t)BATCH * CAND * sizeof(int));
    sp.selIdx[l] = si; dp.selIdx[l] = si;
  }
  for (int l = 0; l < NLVL; ++l) {
    float* sb = (float*)alloc((size_t)BATCH * CAND * 4 * sizeof(float));
    dp.selB[l] = sb; np.selB[l] = sb;
  }
  for (int l = 0; l < NLVL; ++l) {
    float* ss = (float*)alloc((size_t)BATCH * CAND * sizeof(float));
    dp.selS[l] = ss; np.selS[l] = ss;
  }
  float* allB = (float*)alloc((size_t)BATCH * NCAT * 4 * sizeof(float));
  float* allS = (float*)alloc((size_t)BATCH * NCAT * sizeof(float));
  int* finalIdx = (int*)alloc((size_t)BATCH * CAND * sizeof(int));
  np.allB = allB; np.allS = allS;

  // 1) per-(level,batch) exact top-k selection: 40 independent blocks, one launch
  select_topk_fused<<<NLVL * BATCH, 1024, 0, stream>>>(sp);
  // 2) decode+clip+sigmoid for all selected candidates
  decode_fused<<<(dp.ttotal + 255) / 256, 256, 0, stream>>>(dp);
  // 3) greedy NMS for all 40 (level,batch) pairs concurrently
  nms_fused<<<NLVL * BATCH, 512, 0, stream>>>(np);
  // 4) final top-1000 across 4768 concatenated proposals (reuse radix selector)
  select_topk_single<<<BATCH, 1024, 0, stream>>>(allS, NCAT, FINAL_K, finalIdx);
  float* out = (float*)d_out;
  final_gather_kernel<<<(BATCH * FINAL_K + 255) / 256, 256, 0, stream>>>(
      allB, allS, finalIdx, out, out + (size_t)BATCH * FINAL_K * 4);
}